// Attention1_18915035971818
// MI455X (gfx1250) — compile-verified
//
#include <hip/hip_runtime.h>

typedef float v2f __attribute__((ext_vector_type(2)));
typedef float v8f __attribute__((ext_vector_type(8)));

#define WG_M 64
#define WG_N 256
#define KC 16
#define ASTRIDE 18  // pad 64x16 A tile rows: stride 18 floats spreads banks, keeps 8B align

// C[M,N] = A[M,K] * B, where B = Bsrc^T (TRANSB) or Bsrc (!TRANSB).
// 256 threads = 8 wave32 arranged 2(M) x 4(N); each wave owns 2x4 16x16 tiles.
// fp32 matrix math via v_wmma_f32_16x16x4_f32 (exact fp32 — no bf16 demotion).
// Double-buffered LDS: global loads for chunk c+1 are staged in registers
// while WMMAs consume chunk c, then committed before the single barrier.
template <bool TRANSB>
__global__ __launch_bounds__(256) void gemm_wmma_f32(
    const float* __restrict__ A, const float* __restrict__ Bsrc,
    float* __restrict__ C, int K, int lda, int ldb, int ldc,
    long long strideA, long long strideB, long long strideC) {
  __shared__ float As[2][WG_M * ASTRIDE];
  __shared__ float2 Bs[2][(KC / 2) * WG_N];  // K-pairs pre-interleaved -> b64 frag loads

  const int b = blockIdx.z;
  A += (long long)b * strideA;
  Bsrc += (long long)b * strideB;
  C += (long long)b * strideC;

  const int tid = threadIdx.x;
  const int lane = tid & 31;
  const int wid = tid >> 5;
  const int wm = wid >> 2;  // 0..1 -> 32 rows each
  const int wn = wid & 3;   // 0..3 -> 64 cols each

  const int m0 = blockIdx.y * WG_M;
  const int n0 = blockIdx.x * WG_N;

  v8f zero = {};
  v8f acc[2][4];
#pragma unroll
  for (int i = 0; i < 2; ++i)
#pragma unroll
    for (int j = 0; j < 4; ++j) acc[i][j] = zero;

  // --- register staging for the next k-chunk ---
  float4 areg;
  float4 breg[4];

  auto load_stage = [&](int k0) {
    {  // A: 64x16 tile, 1 float4 per thread
      const int row = tid >> 2;
      const int c4 = (tid & 3) << 2;
      areg = *(const float4*)(A + (long long)(m0 + row) * lda + k0 + c4);
    }
    if (TRANSB) {
      // column n = tid of B comes from row (n0+tid) of Bsrc: 16 contiguous floats
      const float* src = Bsrc + (long long)(n0 + tid) * ldb + k0;
#pragma unroll
      for (int i = 0; i < 4; ++i) breg[i] = *(const float4*)(src + 4 * i);
    } else {
      // row-pairs of Bsrc; thread covers 2 groups of 4 columns
#pragma unroll
      for (int r = 0; r < 2; ++r) {
        const int g = tid + 256 * r;  // 0..511
        const int kkp = g >> 6;       // pair-row 0..7
        const int c4 = (g & 63) << 2;
        const float* base = Bsrc + (long long)(k0 + 2 * kkp) * ldb + n0 + c4;
        breg[2 * r] = *(const float4*)(base);
        breg[2 * r + 1] = *(const float4*)(base + ldb);
      }
    }
  };

  auto commit_stage = [&](int buf) {
    {
      const int row = tid >> 2;
      const int c4 = (tid & 3) << 2;
      float* dst = &As[buf][row * ASTRIDE + c4];
      dst[0] = areg.x; dst[1] = areg.y; dst[2] = areg.z; dst[3] = areg.w;
    }
    if (TRANSB) {
#pragma unroll
      for (int i = 0; i < 4; ++i) {
        Bs[buf][(2 * i) * WG_N + tid] = make_float2(breg[i].x, breg[i].y);
        Bs[buf][(2 * i + 1) * WG_N + tid] = make_float2(breg[i].z, breg[i].w);
      }
    } else {
#pragma unroll
      for (int r = 0; r < 2; ++r) {
        const int g = tid + 256 * r;
        const int kkp = g >> 6;
        const int c4 = (g & 63) << 2;
        float2* dst = &Bs[buf][kkp * WG_N + c4];
        const float4 a = breg[2 * r], bb = breg[2 * r + 1];
        dst[0] = make_float2(a.x, bb.x);
        dst[1] = make_float2(a.y, bb.y);
        dst[2] = make_float2(a.z, bb.z);
        dst[3] = make_float2(a.w, bb.w);
      }
    }
  };

  const int arow = wm * 32 + (lane & 15);
  const int koff = (lane >> 4) << 1;  // lanes 16-31 carry K+2,K+3
  const int bcolbase = wn * 64 + (lane & 15);

  load_stage(0);
  commit_stage(0);
  __syncthreads();

  const int nchunks = K / KC;
  for (int c = 0; c < nchunks; ++c) {
    const int cur = c & 1;
    if (c + 1 < nchunks) load_stage((c + 1) * KC);  // overlap global with WMMA

#pragma unroll
    for (int kk = 0; kk < KC; kk += 4) {
      const int ak = kk + koff;  // even
      v2f afrag[2];
#pragma unroll
      for (int i = 0; i < 2; ++i)
        afrag[i] = *(const v2f*)&As[cur][(arow + 16 * i) * ASTRIDE + ak];
      v2f bfrag[4];
#pragma unroll
      for (int j = 0; j < 4; ++j)
        bfrag[j] =
            *(const v2f*)&Bs[cur][(ak >> 1) * WG_N + bcolbase + 16 * j];
#pragma unroll
      for (int i = 0; i < 2; ++i)
#pragma unroll
        for (int j = 0; j < 4; ++j)
          acc[i][j] = __builtin_amdgcn_wmma_f32_16x16x4_f32(
              false, afrag[i], false, bfrag[j], (short)0, acc[i][j], false,
              false);
    }

    if (c + 1 < nchunks) commit_stage(cur ^ 1);
    __syncthreads();
  }

  // ---- epilogue: lane<16: M=v,N=lane ; lane>=16: M=v+8,N=lane-16
  const int crow = m0 + wm * 32 + ((lane >> 4) << 3);
  const int ccol = n0 + wn * 64 + (lane & 15);
#pragma unroll
  for (int i = 0; i < 2; ++i)
#pragma unroll
    for (int j = 0; j < 4; ++j)
#pragma unroll
      for (int v = 0; v < 8; ++v)
        C[(long long)(crow + i * 16 + v) * ldc + ccol + j * 16] = acc[i][j][v];
}

// In-place row softmax over K elements; one 256-thread block per row.
__global__ __launch_bounds__(256) void softmax_rows(float* __restrict__ attn,
                                                    int K) {
  __shared__ float red[256];
  const int tid = threadIdx.x;
  float* row = attn + (long long)blockIdx.x * K;

  float v[8];
  float mx = -3.402823466e38f;
#pragma unroll
  for (int i = 0; i < 8; ++i) {
    v[i] = row[tid + 256 * i];
    mx = fmaxf(mx, v[i]);
  }
  red[tid] = mx;
  __syncthreads();
  for (int s = 128; s > 0; s >>= 1) {
    if (tid < s) red[tid] = fmaxf(red[tid], red[tid + s]);
    __syncthreads();
  }
  mx = red[0];
  __syncthreads();

  float sum = 0.0f;
#pragma unroll
  for (int i = 0; i < 8; ++i) {
    v[i] = __expf(v[i] - mx);
    sum += v[i];
  }
  red[tid] = sum;
  __syncthreads();
  for (int s = 128; s > 0; s >>= 1) {
    if (tid < s) red[tid] += red[tid + s];
    __syncthreads();
  }
  const float inv = 1.0f / red[0];
#pragma unroll
  for (int i = 0; i < 8; ++i) row[tid + 256 * i] = v[i] * inv;
}

extern "C" void kernel_launch(void* const* d_in, const int* in_sizes, int n_in,
                              void* d_out, int out_size, void* d_ws,
                              size_t ws_size, hipStream_t stream) {
  (void)in_sizes; (void)n_in; (void)out_size; (void)d_ws; (void)ws_size;
  const float* dec = (const float*)d_in[0];  // [16,1024,1024]
  const float* enc = (const float*)d_in[1];  // [16,2048,1024]
  float* ctx = (float*)d_out;                              // [16,1024,1024]
  float* attn = (float*)d_out + (size_t)16 * 1024 * 1024;  // [16,1024,2048]

  // Pass 1: raw scores = dec @ enc^T  (per batch 1024x2048, K=1024)
  dim3 g1(2048 / WG_N, 1024 / WG_M, 16);
  gemm_wmma_f32<true><<<g1, 256, 0, stream>>>(
      dec, enc, attn, /*K=*/1024, /*lda=*/1024, /*ldb=*/1024, /*ldc=*/2048,
      1024LL * 1024, 2048LL * 1024, 1024LL * 2048);

  // Pass 2: softmax over last dim, in place (16*1024 rows of 2048)
  softmax_rows<<<16 * 1024, 256, 0, stream>>>(attn, 2048);

  // Pass 3: context = attn @ enc  (per batch 1024x1024, K=2048)
  dim3 g2(1024 / WG_N, 1024 / WG_M, 16);
  gemm_wmma_f32<false><<<g2, 256, 0, stream>>>(
      attn, enc, ctx, /*K=*/2048, /*lda=*/2048, /*ldb=*/1024, /*ldc=*/1024,
      1024LL * 2048, 2048LL * 1024, 1024LL * 1024);
}